// LMI_87789131530707
// MI455X (gfx1250) — compile-verified
//
#include <hip/hip_runtime.h>
#include <math.h>

#define NBINS   32
#define NBINS2  1024          // 32*32
#define MAXPART 64            // partial global histograms (contention spreading)
#define EPSV    1e-5f

typedef __attribute__((ext_vector_type(2))) float v2f;
typedef __attribute__((ext_vector_type(8))) float v8f;

// ---------------------------------------------------------------------------
// Kernel 0: zero the partial-histogram workspace
// ---------------------------------------------------------------------------
__global__ void mi_zero_kernel(unsigned* __restrict__ p, int n) {
    int i = blockIdx.x * blockDim.x + threadIdx.x;
    if (i < n) p[i] = 0u;
}

// ---------------------------------------------------------------------------
// Kernel 1: bandwidth-bound joint histogram.
//   float4 (B128) loads, LDS-privatized 1024-bin histogram (ds_add_u32),
//   global_prefetch a few grid-strides ahead, merge into d_ws partials.
// ---------------------------------------------------------------------------
__global__ void mi_hist_kernel(const float* __restrict__ I,
                               const float* __restrict__ J,
                               unsigned* __restrict__ partial,
                               long long n, long long n4, int npart) {
    __shared__ unsigned lhist[NBINS2];
    for (int i = threadIdx.x; i < NBINS2; i += blockDim.x) lhist[i] = 0u;
    __syncthreads();

    const float4* __restrict__ I4 = (const float4*)I;
    const float4* __restrict__ J4 = (const float4*)J;
    const float s = 31.0f / 255.0f;
    const long long stride = (long long)gridDim.x * blockDim.x;
    long long i = (long long)blockIdx.x * blockDim.x + threadIdx.x;

    for (; i < n4; i += stride) {
        long long pf = i + 4 * stride;           // prefetch ~4 iterations ahead
        if (pf < n4) {
            __builtin_prefetch(&I4[pf], 0, 0);   // -> global_prefetch_b8
            __builtin_prefetch(&J4[pf], 0, 0);
        }
        float4 a = I4[i];
        float4 b = J4[i];
        int i0 = __float2int_rn(a.x * s), j0 = __float2int_rn(b.x * s);
        int i1 = __float2int_rn(a.y * s), j1 = __float2int_rn(b.y * s);
        int i2 = __float2int_rn(a.z * s), j2 = __float2int_rn(b.z * s);
        int i3 = __float2int_rn(a.w * s), j3 = __float2int_rn(b.w * s);
        atomicAdd(&lhist[(i0 << 5) + j0], 1u);   // ds_add_u32
        atomicAdd(&lhist[(i1 << 5) + j1], 1u);
        atomicAdd(&lhist[(i2 << 5) + j2], 1u);
        atomicAdd(&lhist[(i3 << 5) + j3], 1u);
    }
    // scalar tail (n not a multiple of 4) handled by block 0
    if (blockIdx.x == 0) {
        for (long long k = n4 * 4 + threadIdx.x; k < n; k += blockDim.x) {
            int ia = __float2int_rn(I[k] * s);
            int ja = __float2int_rn(J[k] * s);
            atomicAdd(&lhist[(ia << 5) + ja], 1u);
        }
    }
    __syncthreads();

    unsigned* __restrict__ dst = partial + (size_t)(blockIdx.x % npart) * NBINS2;
    for (int k = threadIdx.x; k < NBINS2; k += blockDim.x) {
        unsigned v = lhist[k];
        if (v) atomicAdd(&dst[k], v);            // spread L2 atomics
    }
}

// ---------------------------------------------------------------------------
// Kernel 2: finalize. Sum partials -> P, marginals via V_WMMA_F32_16X16X4_F32
// (rowsum = P x ones, colsum = P^T x ones), MI reduction, sigmoid(-MI).
// One block of 256 threads; wave 0 runs the WMMAs (EXEC all-ones in-wave).
// ---------------------------------------------------------------------------
__global__ void mi_finalize_kernel(const unsigned* __restrict__ partial,
                                   float* __restrict__ out, int npart) {
    __shared__ float Psh[NBINS2];
    __shared__ float Ip[NBINS];
    __shared__ float Jp[NBINS];
    __shared__ float sarr[256];

    const int t = threadIdx.x;

    // --- sum partial histograms: 4 bins per thread ---
    float local = 0.0f;
    float cnt[4];
    for (int k = 0; k < 4; ++k) {
        int bin = t * 4 + k;
        unsigned s = 0u;
        for (int p = 0; p < npart; ++p) s += partial[(size_t)p * NBINS2 + bin];
        cnt[k] = (float)s;
        local += cnt[k];
    }
    // deterministic tree reduction for total count
    sarr[t] = local;
    __syncthreads();
    for (int off = 128; off > 0; off >>= 1) {
        if (t < off) sarr[t] += sarr[t + off];
        __syncthreads();
    }
    const float inv = 1.0f / sarr[0];
    __syncthreads();

    for (int k = 0; k < 4; ++k) Psh[t * 4 + k] = cnt[k] * inv;
    __syncthreads();

    // --- marginals via WMMA (wave 0 only; full-wave EXEC) ---
    if (t < 32) {
        const int lane = t;
        const int mloc = lane & 15;            // A-matrix row (ISA layout)
        const int ks   = (lane >> 4) << 1;     // K pair: 0 for lanes 0-15, 2 for 16-31
        const v2f onesb = {1.0f, 1.0f};        // B = ones: layout-invariant
        for (int half = 0; half < 2; ++half) {
            const int m = mloc + half * 16;
            v8f accR = {};                     // row sums of P (I_prob)
            v8f accC = {};                     // col sums of P (J_prob) = rowsums of P^T
            for (int g = 0; g < 8; ++g) {      // 8 K-groups of 4 cover all 32 columns
                const int c = g * 4 + ks;
                v2f ar, ac;
                ar.x = Psh[m * NBINS + c];
                ar.y = Psh[m * NBINS + c + 1];
                ac.x = Psh[c * NBINS + m];
                ac.y = Psh[(c + 1) * NBINS + m];
                accR = __builtin_amdgcn_wmma_f32_16x16x4_f32(
                           false, ar, false, onesb, (short)0, accR, false, false);
                accC = __builtin_amdgcn_wmma_f32_16x16x4_f32(
                           false, ac, false, onesb, (short)0, accC, false, false);
            }
            // D layout: lane<16 holds M=0..7 in v0..v7 (N=lane); lane>=16 holds M=8..15.
            const int mbase = half * 16 + ((lane >= 16) ? 8 : 0);
            if (mloc == 0) {                   // lanes 0 and 16 extract (all N identical)
                for (int r = 0; r < 8; ++r) {
                    Ip[mbase + r] = accR[r];
                    Jp[mbase + r] = accC[r];
                }
            }
        }
    }
    __syncthreads();

    // --- MI accumulation: 4 bins per thread, deterministic tree reduce ---
    float mi = 0.0f;
    for (int k = 0; k < 4; ++k) {
        int bin = t * 4 + k;
        int i = bin >> 5, j = bin & 31;
        float p = Psh[bin];
        mi += p * (logf(p + EPSV) - logf(Ip[i] + EPSV) - logf(Jp[j] + EPSV));
    }
    sarr[t] = mi;
    __syncthreads();
    for (int off = 128; off > 0; off >>= 1) {
        if (t < off) sarr[t] += sarr[t + off];
        __syncthreads();
    }
    if (t == 0) {
        float m = sarr[0];
        out[0] = 1.0f / (1.0f + expf(m));      // sigmoid(-MI)
    }
}

// ---------------------------------------------------------------------------
extern "C" void kernel_launch(void* const* d_in, const int* in_sizes, int n_in,
                              void* d_out, int out_size, void* d_ws, size_t ws_size,
                              hipStream_t stream) {
    const float* I = (const float*)d_in[0];
    const float* J = (const float*)d_in[1];
    unsigned* partial = (unsigned*)d_ws;
    float* out = (float*)d_out;

    const long long n  = (long long)in_sizes[0];
    const long long n4 = n >> 2;

    int npart = (int)(ws_size / (NBINS2 * sizeof(unsigned)));
    if (npart > MAXPART) npart = MAXPART;
    if (npart < 1) npart = 1;

    const int nz = npart * NBINS2;
    mi_zero_kernel<<<(nz + 255) / 256, 256, 0, stream>>>(partial, nz);

    const int blocks = 2048;                   // grid-stride; saturates HBM
    mi_hist_kernel<<<blocks, 256, 0, stream>>>(I, J, partial, n, n4, npart);

    mi_finalize_kernel<<<1, 256, 0, stream>>>(partial, out, npart);
}